// DGCF_60035052863933
// MI455X (gfx1250) — compile-verified
//
#include <hip/hip_runtime.h>
#include <stdint.h>

// ---------------------------------------------------------------------------
// DGCF forward for MI455X (gfx1250, wave32).
// Memory-bound gather/scatter pipeline, L2-resident (~120MB < 192MB L2).
// CDNA5 paths: V_WMMA_F32_16X16X4_F32 row-sum reductions (exact f32, ones-B),
// async global->LDS edge staging (ASYNCcnt), wave32 shuffle reductions.
// ---------------------------------------------------------------------------

#define EPSF 1e-12f
#define EB 256  // edges staged per block in spmm

typedef __attribute__((ext_vector_type(2))) float v2f;
typedef __attribute__((ext_vector_type(8))) float v8f;
typedef int v4i32 __attribute__((vector_size(16)));  // matches builtin param

#ifndef __has_builtin
#define __has_builtin(x) 0
#endif

#if __has_builtin(__builtin_amdgcn_wmma_f32_16x16x4_f32)
#define HAVE_WMMA4 1
#else
#define HAVE_WMMA4 0
#endif

#if __has_builtin(__builtin_amdgcn_global_load_async_to_lds_b32)
#define HAVE_ASYNC 1
#else
#define HAVE_ASYNC 0
#endif

#if __has_builtin(__builtin_amdgcn_global_load_async_to_lds_b128)
#define HAVE_ASYNC128 1
#else
#define HAVE_ASYNC128 0
#endif

#define AS1 __attribute__((address_space(1)))
#define AS3 __attribute__((address_space(3)))

// ---------------------------------------------------------------------------
// init: ego = concat(user,item); acc = ego; A_values = 1
__global__ void k_init(const float* __restrict__ eu, const float* __restrict__ ei,
                       float* __restrict__ ego, float* __restrict__ acc,
                       float* __restrict__ Av, int nuel, int totel, int ael) {
  int i = blockIdx.x * blockDim.x + threadIdx.x;
  if (i < totel) {
    float v = (i < nuel) ? eu[i] : ei[i - nuel];
    ego[i] = v;
    acc[i] = v;
  }
  if (i < ael) Av[i] = 1.0f;
}

__global__ void k_zero(float* __restrict__ p, int n) {
  int i = blockIdx.x * blockDim.x + threadIdx.x;
  if (i < n) p[i] = 0.0f;
}

// ---------------------------------------------------------------------------
// per-edge softmax over 4 factors + atomic row-sum accumulation (by head)
__global__ void k_softmax_rowsum(const float4* __restrict__ Av,
                                 const int* __restrict__ head,
                                 float4* __restrict__ vals,
                                 float* __restrict__ rowsum, int nnz) {
  int e = blockIdx.x * blockDim.x + threadIdx.x;
  if (e >= nnz) return;
  float4 a = Av[e];
  float mx = fmaxf(fmaxf(a.x, a.y), fmaxf(a.z, a.w));
  float e0 = __expf(a.x - mx), e1 = __expf(a.y - mx);
  float e2 = __expf(a.z - mx), e3 = __expf(a.w - mx);
  float inv = 1.0f / (e0 + e1 + e2 + e3);
  float4 v = make_float4(e0 * inv, e1 * inv, e2 * inv, e3 * inv);
  vals[e] = v;
  float* rs = rowsum + (size_t)head[e] * 4;
  atomicAdd(rs + 0, v.x);
  atomicAdd(rs + 1, v.y);
  atomicAdd(rs + 2, v.z);
  atomicAdd(rs + 3, v.w);
}

// d = rowsum>0 ? rsqrt(max(rowsum,eps)) : 0
__global__ void k_dcalc(const float* __restrict__ rowsum, float* __restrict__ dmat, int n4) {
  int i = blockIdx.x * blockDim.x + threadIdx.x;
  if (i >= n4) return;
  float r = rowsum[i];
  dmat[i] = (r > 0.0f) ? rsqrtf(fmaxf(r, EPSF)) : 0.0f;
}

// ---------------------------------------------------------------------------
// SpMM: y[head] += vals[f]*d[tail,f]*ego[tail]  (wave per edge, float2/lane)
// Edge streams staged to LDS with CDNA5 async loads (ASYNCcnt path).
__global__ __launch_bounds__(256) void k_spmm(
    const int* __restrict__ head, const int* __restrict__ tail,
    const float4* __restrict__ vals, const float* __restrict__ dmat,
    const float* __restrict__ ego, float* __restrict__ y, int nnz) {
  __shared__ int shh[EB];
  __shared__ int sht[EB];
  __shared__ float shv[EB * 4];
  const int t = threadIdx.x;
  const int base = blockIdx.x * EB;
#if HAVE_ASYNC
  if (base + t < nnz) {
    __builtin_amdgcn_global_load_async_to_lds_b32(
        (AS1 int*)(head + base + t), (AS3 int*)&shh[t], 0, 0);
    __builtin_amdgcn_global_load_async_to_lds_b32(
        (AS1 int*)(tail + base + t), (AS3 int*)&sht[t], 0, 0);
#if HAVE_ASYNC128
    __builtin_amdgcn_global_load_async_to_lds_b128(
        (AS1 v4i32*)(vals + base + t), (AS3 v4i32*)&shv[4 * t], 0, 0);
#else
    *(float4*)&shv[4 * t] = vals[base + t];
#endif
  }
  asm volatile("s_wait_asynccnt 0" ::: "memory");
  __syncthreads();
#else
  if (base + t < nnz) {
    shh[t] = head[base + t];
    sht[t] = tail[base + t];
    *(float4*)&shv[4 * t] = vals[base + t];
  }
  __syncthreads();
#endif
  const int lane = t & 31;
  const int wv = t >> 5;
  const int f = lane >> 3;  // factor owning dims [f*16, f*16+16)
  const int iend = min(EB, nnz - base);
  const int ihi = min((wv + 1) * 32, iend);
  for (int i = wv * 32; i < ihi; ++i) {
    const int hh = shh[i];
    const int tt = sht[i];
    const float coef = shv[4 * i + f] * dmat[tt * 4 + f];
    const float2 x = *(const float2*)(ego + (size_t)tt * 64 + lane * 2);
    float* yr = y + (size_t)hh * 64 + lane * 2;
    atomicAdd(yr + 0, coef * x.x);
    atomicAdd(yr + 1, coef * x.y);
  }
}

// factor_emb = d * y   (in place on y)
__global__ void k_scale(float* __restrict__ y, const float* __restrict__ dmat, int tot) {
  int i = blockIdx.x * blockDim.x + threadIdx.x;
  if (i >= tot) return;
  int node = i >> 6;
  int f = (i >> 4) & 3;
  y[i] *= dmat[node * 4 + f];
}

// ---------------------------------------------------------------------------
#if HAVE_WMMA4
// Row-sum over 16 dims for a 16-edge tile, via 4 chained F32 16x16x4 WMMAs
// with an all-ones B (layout-invariant): D[m,n] = sum_k A[m,k].
static __device__ __forceinline__ v8f rowsum_wmma(v2f a0, v2f a1, v2f a2, v2f a3) {
  const v2f one2 = {1.0f, 1.0f};
  v8f c = {};
  c = __builtin_amdgcn_wmma_f32_16x16x4_f32(false, a0, false, one2, (short)0, c, false, false);
  c = __builtin_amdgcn_wmma_f32_16x16x4_f32(false, a1, false, one2, (short)0, c, false, false);
  c = __builtin_amdgcn_wmma_f32_16x16x4_f32(false, a2, false, one2, (short)0, c, false, false);
  c = __builtin_amdgcn_wmma_f32_16x16x4_f32(false, a3, false, one2, (short)0, c, false, false);
  return c;
}
// D layout (doc 7.12.2): lanes0-15 reg r holds M=r; lanes16-31 reg r holds M=r+8.
static __device__ __forceinline__ void bounce_rows(volatile float* bw, int off, v8f d, int lane) {
  if (lane == 0) {
    bw[off + 0] = d[0]; bw[off + 1] = d[1]; bw[off + 2] = d[2]; bw[off + 3] = d[3];
    bw[off + 4] = d[4]; bw[off + 5] = d[5]; bw[off + 6] = d[6]; bw[off + 7] = d[7];
  } else if (lane == 16) {
    bw[off + 8]  = d[0]; bw[off + 9]  = d[1]; bw[off + 10] = d[2]; bw[off + 11] = d[3];
    bw[off + 12] = d[4]; bw[off + 13] = d[5]; bw[off + 14] = d[6]; bw[off + 15] = d[7];
  }
}
#endif

// Routing: A[e,f] += dot(l2norm(femb[head,f,:]), tanh(l2norm(ego[tail,f,:])))
// Wave handles 16 edges; lane = hi*16 + m owns edge m, dim-pairs hi*2+4*kc
// == exact 16x4 f32 A-matrix WMMA layout (doc 7.12.2).
__global__ __launch_bounds__(256) void k_routing(
    const int* __restrict__ head, const int* __restrict__ tail,
    const float* __restrict__ femb, const float* __restrict__ ego,
    float* __restrict__ Av, int nnz) {
#if HAVE_WMMA4
  __shared__ float bounce[8][48];
#endif
  const int lane = threadIdx.x & 31;
  const int wv = threadIdx.x >> 5;
  const int te = (blockIdx.x * 8 + wv) * 16;  // wave-uniform
  if (te >= nnz) return;
  const int m = lane & 15;
  const int hi = lane >> 4;
  const int e = te + m;
  const int ec = (e < nnz) ? e : (nnz - 1);
  const int hh = head[ec];
  const int tt = tail[ec];
  const float* hrow = femb + (size_t)hh * 64;
  const float* trow = ego + (size_t)tt * 64;
#if HAVE_WMMA4
  volatile float* bw = bounce[wv];
#endif
  for (int f = 0; f < 4; ++f) {
    v2f hv[4], tv[4];
#pragma unroll
    for (int kc = 0; kc < 4; ++kc) {
      const int dd = f * 16 + kc * 4 + hi * 2;
      hv[kc] = *(const v2f*)(hrow + dd);
      tv[kc] = *(const v2f*)(trow + dd);
    }
#if HAVE_WMMA4
    v8f dh = rowsum_wmma(hv[0] * hv[0], hv[1] * hv[1], hv[2] * hv[2], hv[3] * hv[3]);
    v8f dt = rowsum_wmma(tv[0] * tv[0], tv[1] * tv[1], tv[2] * tv[2], tv[3] * tv[3]);
    bounce_rows(bw, 0, dh, lane);
    bounce_rows(bw, 16, dt, lane);
    asm volatile("s_wait_dscnt 0" ::: "memory");
    const float invnh = 1.0f / fmaxf(sqrtf(bw[m]), EPSF);
    const float invnt = 1.0f / fmaxf(sqrtf(bw[16 + m]), EPSF);
    v2f p[4];
#pragma unroll
    for (int kc = 0; kc < 4; ++kc) {
      p[kc].x = hv[kc].x * invnh * tanhf(tv[kc].x * invnt);
      p[kc].y = hv[kc].y * invnh * tanhf(tv[kc].y * invnt);
    }
    v8f dsc = rowsum_wmma(p[0], p[1], p[2], p[3]);
    bounce_rows(bw, 32, dsc, lane);
    asm volatile("s_wait_dscnt 0" ::: "memory");
    if (hi == 0 && e < nnz) Av[4 * e + f] += bw[32 + m];
#else
    float sh2 = 0.0f, st2 = 0.0f;
#pragma unroll
    for (int kc = 0; kc < 4; ++kc) {
      sh2 += hv[kc].x * hv[kc].x + hv[kc].y * hv[kc].y;
      st2 += tv[kc].x * tv[kc].x + tv[kc].y * tv[kc].y;
    }
    sh2 += __shfl_xor(sh2, 16);
    st2 += __shfl_xor(st2, 16);
    const float invnh = 1.0f / fmaxf(sqrtf(sh2), EPSF);
    const float invnt = 1.0f / fmaxf(sqrtf(st2), EPSF);
    float sc = 0.0f;
#pragma unroll
    for (int kc = 0; kc < 4; ++kc) {
      sc += hv[kc].x * invnh * tanhf(tv[kc].x * invnt) +
            hv[kc].y * invnh * tanhf(tv[kc].y * invnt);
    }
    sc += __shfl_xor(sc, 16);
    if (hi == 0 && e < nnz) Av[4 * e + f] += sc;
#endif
  }
}

// ---------------------------------------------------------------------------
// layer end: ego = per-factor l2norm(femb); acc += ego   (wave per node)
__global__ __launch_bounds__(256) void k_layernorm(
    const float* __restrict__ femb, float* __restrict__ ego,
    float* __restrict__ acc, int n) {
  const int lane = threadIdx.x & 31;
  const int node = blockIdx.x * 8 + (threadIdx.x >> 5);
  if (node >= n) return;
  const float2 x = *(const float2*)(femb + (size_t)node * 64 + lane * 2);
  float s2 = x.x * x.x + x.y * x.y;  // 8-lane group == one 16-dim factor
  s2 += __shfl_xor(s2, 1);
  s2 += __shfl_xor(s2, 2);
  s2 += __shfl_xor(s2, 4);
  const float inv = 1.0f / fmaxf(sqrtf(s2), EPSF);
  float2 o;
  o.x = x.x * inv;
  o.y = x.y * inv;
  *(float2*)(ego + (size_t)node * 64 + lane * 2) = o;
  float* ar = acc + (size_t)node * 64 + lane * 2;
  ar[0] += o.x;
  ar[1] += o.y;
}

__global__ void k_final(const float* __restrict__ acc, float* __restrict__ out, int n) {
  int i = blockIdx.x * blockDim.x + threadIdx.x;
  if (i < n) out[i] = acc[i] * (1.0f / 3.0f);
}

// ---------------------------------------------------------------------------
extern "C" void kernel_launch(void* const* d_in, const int* in_sizes, int n_in,
                              void* d_out, int out_size, void* d_ws, size_t ws_size,
                              hipStream_t stream) {
  (void)n_in;
  (void)ws_size;
  const float* eu = (const float*)d_in[0];
  const float* ei = (const float*)d_in[1];
  const int* head = (const int*)d_in[2];
  const int* tail = (const int*)d_in[3];
  const int nuel = in_sizes[0];
  const int niel = in_sizes[1];
  const int nnz = in_sizes[2];
  const int totel = nuel + niel;  // N * 64
  const int N = totel / 64;

  char* wsb = (char*)d_ws;
  size_t off = 0;
  auto alloc = [&](size_t bytes) -> float* {
    size_t a = (off + 255) & ~(size_t)255;
    off = a + bytes;
    return (float*)(wsb + a);
  };
  float* ego = alloc((size_t)totel * 4);
  float* acc = alloc((size_t)totel * 4);
  float* y = alloc((size_t)totel * 4);
  float* Av = alloc((size_t)nnz * 16);
  float* vals = alloc((size_t)nnz * 16);
  float* rowsum = alloc((size_t)N * 16);
  float* dmat = alloc((size_t)N * 16);

  const int B = 256;
  const int initMax = (totel > nnz * 4) ? totel : nnz * 4;
  const int gInit = (initMax + B - 1) / B;
  const int gE = (nnz + B - 1) / B;
  const int gN4 = (N * 4 + B - 1) / B;
  const int gTot = (totel + B - 1) / B;
  const int gSp = (nnz + EB - 1) / EB;
  const int gRt = ((nnz + 15) / 16 + 7) / 8;
  const int gLn = (N + 7) / 8;

  k_init<<<gInit, B, 0, stream>>>(eu, ei, ego, acc, Av, nuel, totel, nnz * 4);

  for (int layer = 0; layer < 2; ++layer) {
    for (int it = 0; it < 2; ++it) {
      k_zero<<<gN4, B, 0, stream>>>(rowsum, N * 4);
      k_softmax_rowsum<<<gE, B, 0, stream>>>((const float4*)Av, head, (float4*)vals,
                                             rowsum, nnz);
      k_dcalc<<<gN4, B, 0, stream>>>(rowsum, dmat, N * 4);
      k_zero<<<gTot, B, 0, stream>>>(y, totel);
      k_spmm<<<gSp, B, 0, stream>>>(head, tail, (const float4*)vals, dmat, ego, y, nnz);
      k_scale<<<gTot, B, 0, stream>>>(y, dmat, totel);
      // final routing update of the last iteration is never consumed -> skip
      if (!(layer == 1 && it == 1))
        k_routing<<<gRt, B, 0, stream>>>(head, tail, y, ego, Av, nnz);
    }
    k_layernorm<<<gLn, B, 0, stream>>>(y, ego, acc, N);
  }
  k_final<<<(out_size + B - 1) / B, B, 0, stream>>>(acc, (float*)d_out, out_size);
}